// SSingleMatchNet_36197984371338
// MI455X (gfx1250) — compile-verified
//
#include <hip/hip_runtime.h>

typedef float v2f __attribute__((ext_vector_type(2)));
typedef float v4f __attribute__((ext_vector_type(4)));
typedef float v8f __attribute__((ext_vector_type(8)));
typedef unsigned int u32x4 __attribute__((ext_vector_type(4)));
typedef int i32x4 __attribute__((ext_vector_type(4)));
typedef int i32x8 __attribute__((ext_vector_type(8)));

#define B_  16
#define LP_ 512
#define LQ_ 512
#define H_  768

#define WMMA_F32(a, b, c) \
  __builtin_amdgcn_wmma_f32_16x16x4_f32(false, (a), false, (b), (short)0, (c), false, false)

#if __has_builtin(__builtin_amdgcn_tensor_load_to_lds) && \
    __has_builtin(__builtin_amdgcn_s_wait_tensorcnt)
#define USE_TDM 1
#else
#define USE_TDM 0
#endif

#define KT      32            // K panel
#define LDSTR   36            // padded LDS row stride (floats): 32 + 4

#if USE_TDM
// Issue TENSOR_LOAD_TO_LDS of a 2D tile [tile_d1 rows x KT cols] (fp32),
// row stride `stride0` elements, into LDS at byte offset `lds_byte`,
// with +4-DWORD padding every 32 DWORDs (-> stride-36 rows in LDS).
__device__ __forceinline__ void tdm_load_panel(const float* gsrc,
                                               unsigned int lds_byte,
                                               unsigned int tensor_d0,
                                               unsigned int tensor_d1,
                                               unsigned long long stride0,
                                               unsigned int tile_d1) {
  const unsigned long long ga = (unsigned long long)(size_t)gsrc;
  u32x4 g0;
  g0.x = 1u;                                            // count=1 (valid), user mode
  g0.y = lds_byte;                                      // lds_addr
  g0.z = (unsigned int)(ga & 0xffffffffull);            // global_addr[31:0]
  g0.w = (unsigned int)((ga >> 32) & 0x01ffffffull)     // global_addr[56:32]
         | 0x80000000u;                                 // type=2 ("image")
  i32x8 g1;
  g1[0] = (int)((2u << 16)        // data_size = 4 bytes
              | (1u << 20)        // pad_enable
              | (4u << 22)        // pad_interval: 32 DWORDs
              | (3u << 25));      // pad_amount:   4 DWORDs
  g1[1] = (int)((tensor_d0 & 0xffffu) << 16);           // abar=0 | td0[15:0]
  g1[2] = (int)((tensor_d0 >> 16) | ((tensor_d1 & 0xffffu) << 16));
  g1[3] = (int)((tensor_d1 >> 16) | ((unsigned)KT << 16));   // | tile_dim0
  g1[4] = (int)(tile_d1 & 0xffffu);                     // tile_dim1 | tile_dim2=0
  g1[5] = (int)(unsigned int)(stride0 & 0xffffffffull); // stride0[31:0]
  g1[6] = (int)(unsigned int)((stride0 >> 32) & 0xffffull); // stride0[47:32] | stride1=0
  g1[7] = 0;
  i32x4 gz4 = {0, 0, 0, 0};
  i32x8 gz8 = {0, 0, 0, 0, 0, 0, 0, 0};
  __builtin_amdgcn_tensor_load_to_lds(g0, g1, gz4, gz4, gz8, 0);
}
#endif

// ---------------------------------------------------------------------------
// C[m,n] = act( sum_k A[m,k]*Bm[n,k] + bias[n] )   (A: MxK, Bm: NxK)
// Block tile 64x128, K-panels of 32 double-buffered in LDS (TDM DMA when
// available). 8 waves, each computing a 16x64 strip (4 accumulators) from
// ds_load_b64 fragments. Requires M%64==0, N%128==0, K%32==0.
// ---------------------------------------------------------------------------
__global__ void gemm_nt_bias_kernel(const float* __restrict__ A,
                                    const float* __restrict__ Bm,
                                    const float* __restrict__ bias,
                                    float* __restrict__ C,
                                    int M, int N, int K, int relu) {
  __shared__ float As[2][64 * LDSTR];    // 2 x 9216 B
  __shared__ float Bs[2][128 * LDSTR];   // 2 x 18432 B

  const int lane = threadIdx.x & 31;
  const int wave = threadIdx.x >> 5;
  const int half = lane >> 4;
  const int idx  = lane & 15;
  const int kb   = half * 2;

  const int nb = N >> 7;                 // 128-wide column blocks
  const int mt = blockIdx.x / nb;        // 64-row block
  const int nt = blockIdx.x % nb;
  const int m0 = mt * 64;
  const int n0 = nt * 128;

  const int m_local = (wave >> 1) * 16;  // 4 m-tiles x 2 n-groups
  const int n_local = (wave & 1) * 64;

  const int panels = K / KT;

#if USE_TDM
  if (wave == 0) {
    tdm_load_panel(A  + (size_t)m0 * K, (unsigned int)(size_t)&As[0][0],
                   (unsigned int)K, 64u, (unsigned long long)K, 64u);
    tdm_load_panel(Bm + (size_t)n0 * K, (unsigned int)(size_t)&Bs[0][0],
                   (unsigned int)K, 128u, (unsigned long long)K, 128u);
    __builtin_amdgcn_s_wait_tensorcnt(0);
  }
  __syncthreads();
#else
  {
    const int t = threadIdx.x;
    for (int c = t; c < 512; c += 256) {         // A: 64 rows x 8 float4
      const int row = c >> 3, col = (c & 7) * 4;
      *(v4f*)&As[0][row * LDSTR + col] =
          *(const v4f*)&A[(size_t)(m0 + row) * K + col];
    }
    for (int c = t; c < 1024; c += 256) {        // B: 128 rows x 8 float4
      const int row = c >> 3, col = (c & 7) * 4;
      *(v4f*)&Bs[0][row * LDSTR + col] =
          *(const v4f*)&Bm[(size_t)(n0 + row) * K + col];
    }
  }
  __syncthreads();
#endif

  v8f acc0 = {}, acc1 = {}, acc2 = {}, acc3 = {};
  int buf = 0;

  for (int p = 0; p < panels; ++p) {
    // start DMA of next panel into the other buffer while we compute
    if (p + 1 < panels) {
#if USE_TDM
      if (wave == 0) {
        const int kofs = (p + 1) * KT;
        tdm_load_panel(A  + (size_t)m0 * K + kofs,
                       (unsigned int)(size_t)&As[buf ^ 1][0],
                       (unsigned int)K, 64u, (unsigned long long)K, 64u);
        tdm_load_panel(Bm + (size_t)n0 * K + kofs,
                       (unsigned int)(size_t)&Bs[buf ^ 1][0],
                       (unsigned int)K, 128u, (unsigned long long)K, 128u);
      }
#else
      const int t = threadIdx.x;
      const int kofs = (p + 1) * KT;
      for (int c = t; c < 512; c += 256) {
        const int row = c >> 3, col = (c & 7) * 4;
        *(v4f*)&As[buf ^ 1][row * LDSTR + col] =
            *(const v4f*)&A[(size_t)(m0 + row) * K + kofs + col];
      }
      for (int c = t; c < 1024; c += 256) {
        const int row = c >> 3, col = (c & 7) * 4;
        *(v4f*)&Bs[buf ^ 1][row * LDSTR + col] =
            *(const v4f*)&Bm[(size_t)(n0 + row) * K + kofs + col];
      }
#endif
    }

    // compute this panel from LDS
    const float* aB = &As[buf][(m_local + idx) * LDSTR + kb];
    const float* b0 = &Bs[buf][(n_local + idx) * LDSTR + kb];
    const float* b1 = b0 + 16 * LDSTR;
    const float* b2 = b0 + 32 * LDSTR;
    const float* b3 = b0 + 48 * LDSTR;
#pragma unroll
    for (int k = 0; k < KT; k += 4) {
      v2f a  = *(const v2f*)(aB + k);
      v2f f0 = *(const v2f*)(b0 + k);
      v2f f1 = *(const v2f*)(b1 + k);
      v2f f2 = *(const v2f*)(b2 + k);
      v2f f3 = *(const v2f*)(b3 + k);
      acc0 = WMMA_F32(a, f0, acc0);
      acc1 = WMMA_F32(a, f1, acc1);
      acc2 = WMMA_F32(a, f2, acc2);
      acc3 = WMMA_F32(a, f3, acc3);
    }

#if USE_TDM
    if (wave == 0) __builtin_amdgcn_s_wait_tensorcnt(0);
#endif
    __syncthreads();
    buf ^= 1;
  }

  const int mrow = m0 + m_local + half * 8;
#pragma unroll
  for (int j = 0; j < 4; ++j) {
    const v8f acc = (j == 0) ? acc0 : (j == 1) ? acc1 : (j == 2) ? acc2 : acc3;
    const int n   = n0 + n_local + j * 16 + idx;
    const float bv = bias[n];
    float* Cb = C + (size_t)mrow * N + n;
#pragma unroll
    for (int r = 0; r < 8; ++r) {
      float v = acc[r] + bv;
      if (relu) v = fmaxf(v, 0.0f);
      Cb[(size_t)r * N] = v;
    }
  }
}

// ---------------------------------------------------------------------------
// Fused attention: per block = (batch b, 32 p-rows), 8 waves. (unchanged)
// ---------------------------------------------------------------------------
__global__ void attn_softmax_kernel(const float* __restrict__ proj_p,
                                    const float* __restrict__ proj_q,
                                    const float* __restrict__ trans_q,
                                    const int*   __restrict__ seq_len,
                                    float* __restrict__ att_vec) {
  extern __shared__ float S[];                  // [32][LQ_] = 64 KB
  const int b  = blockIdx.y;
  const int p0 = blockIdx.x * 32;

  const float* P  = proj_p  + (size_t)b * LP_ * H_;
  const float* Q  = proj_q  + (size_t)b * LQ_ * H_;
  const float* TQ = trans_q + (size_t)b * LQ_ * H_;

  const int lane = threadIdx.x & 31;
  const int wave = threadIdx.x >> 5;
  const int half = lane >> 4;
  const int idx  = lane & 15;
  const int kb   = half * 2;

  // ---- Phase A: S[p][q] = sum_h P[p0+p][h] * TQ[q][h] ----
  for (int sp = wave; sp < 16; sp += 8) {
    const int pt = sp >> 3;
    const int q0 = (sp & 7) * 64;
    const float* Arow = P  + (size_t)(p0 + pt * 16 + idx) * H_ + kb;
    const float* B0   = TQ + (size_t)(q0 + idx) * H_ + kb;
    const float* B1   = B0 + (size_t)16 * H_;
    const float* B2   = B0 + (size_t)32 * H_;
    const float* B3   = B0 + (size_t)48 * H_;

    v8f acc0 = {}, acc1 = {}, acc2 = {}, acc3 = {};
    v2f aA  = *(const v2f*)(Arow);
    v2f b0A = *(const v2f*)(B0);
    v2f b1A = *(const v2f*)(B1);
    v2f b2A = *(const v2f*)(B2);
    v2f b3A = *(const v2f*)(B3);
    v2f aB  = *(const v2f*)(Arow + 4);
    v2f b0B = *(const v2f*)(B0 + 4);
    v2f b1B = *(const v2f*)(B1 + 4);
    v2f b2B = *(const v2f*)(B2 + 4);
    v2f b3B = *(const v2f*)(B3 + 4);

    for (int k = 0; k <= H_ - 16; k += 8) {
      v2f a = aA, b0 = b0A, b1 = b1A, b2 = b2A, b3 = b3A;
      aA  = *(const v2f*)(Arow + k + 8);
      b0A = *(const v2f*)(B0 + k + 8);
      b1A = *(const v2f*)(B1 + k + 8);
      b2A = *(const v2f*)(B2 + k + 8);
      b3A = *(const v2f*)(B3 + k + 8);
      acc0 = WMMA_F32(a, b0, acc0);
      acc1 = WMMA_F32(a, b1, acc1);
      acc2 = WMMA_F32(a, b2, acc2);
      acc3 = WMMA_F32(a, b3, acc3);
      v2f a_ = aB, c0 = b0B, c1 = b1B, c2 = b2B, c3 = b3B;
      aB  = *(const v2f*)(Arow + k + 12);
      b0B = *(const v2f*)(B0 + k + 12);
      b1B = *(const v2f*)(B1 + k + 12);
      b2B = *(const v2f*)(B2 + k + 12);
      b3B = *(const v2f*)(B3 + k + 12);
      acc0 = WMMA_F32(a_, c0, acc0);
      acc1 = WMMA_F32(a_, c1, acc1);
      acc2 = WMMA_F32(a_, c2, acc2);
      acc3 = WMMA_F32(a_, c3, acc3);
    }
    acc0 = WMMA_F32(aA, b0A, acc0);
    acc1 = WMMA_F32(aA, b1A, acc1);
    acc2 = WMMA_F32(aA, b2A, acc2);
    acc3 = WMMA_F32(aA, b3A, acc3);
    acc0 = WMMA_F32(aB, b0B, acc0);
    acc1 = WMMA_F32(aB, b1B, acc1);
    acc2 = WMMA_F32(aB, b2B, acc2);
    acc3 = WMMA_F32(aB, b3B, acc3);

    float* Srow = S + (size_t)(pt * 16 + half * 8) * LQ_ + idx;
#pragma unroll
    for (int j = 0; j < 4; ++j) {
      const v8f acc = (j == 0) ? acc0 : (j == 1) ? acc1 : (j == 2) ? acc2 : acc3;
      float* Sp = Srow + q0 + j * 16;
#pragma unroll
      for (int r = 0; r < 8; ++r) Sp[(size_t)r * LQ_] = acc[r];
    }
  }
  __syncthreads();

  // ---- Phase B: masked softmax, 4 rows per wave ----
  const int L = seq_len[b];
  for (int r = wave * 4; r < wave * 4 + 4; ++r) {
    float* row = S + (size_t)r * LQ_;
    float mx = -3.4e38f;
#pragma unroll
    for (int j = 0; j < LQ_ / 32; ++j) {
      const int q = lane + 32 * j;
      if (q < L) mx = fmaxf(mx, row[q]);
    }
    for (int o = 16; o > 0; o >>= 1) mx = fmaxf(mx, __shfl_xor(mx, o, 32));

    float vals[LQ_ / 32];
    float sum = 0.0f;
#pragma unroll
    for (int j = 0; j < LQ_ / 32; ++j) {
      const int q = lane + 32 * j;
      const float e = (q < L) ? __expf(row[q] - mx) : 0.0f;
      vals[j] = e;
      sum += e;
    }
    for (int o = 16; o > 0; o >>= 1) sum += __shfl_xor(sum, o, 32);
    const float inv = 1.0f / (sum + 1e-13f);
#pragma unroll
    for (int j = 0; j < LQ_ / 32; ++j) row[lane + 32 * j] = vals[j] * inv;
  }
  __syncthreads();

  // ---- Phase C: att_vec[p][h] = sum_q S[p][q] * Q[q][h] ----
  float* AV = att_vec + ((size_t)b * LP_ + p0) * H_;
  for (int sp = wave; sp < 24; sp += 8) {
    const int pt = sp / 12;
    const int n0 = (sp % 12) * 64;
    const float* Srow = S + (size_t)(pt * 16 + idx) * LQ_ + kb;
    const float* Qc0  = Q + (size_t)kb * H_ + n0 + idx;
    const float* Qc1  = Qc0 + 16;
    const float* Qc2  = Qc0 + 32;
    const float* Qc3  = Qc0 + 48;

#define LDB_C(kk, av, v0, v1, v2, v3)                                  \
    {                                                                  \
      const size_t r0 = (size_t)(kk)*H_, r1 = (size_t)((kk) + 1) * H_; \
      av = *(const v2f*)(Srow + (kk));                                 \
      v0.x = Qc0[r0]; v0.y = Qc0[r1];                                  \
      v1.x = Qc1[r0]; v1.y = Qc1[r1];                                  \
      v2.x = Qc2[r0]; v2.y = Qc2[r1];                                  \
      v3.x = Qc3[r0]; v3.y = Qc3[r1];                                  \
    }

    v8f acc0 = {}, acc1 = {}, acc2 = {}, acc3 = {};
    v2f aA, b0A, b1A, b2A, b3A, aB, b0B, b1B, b2B, b3B;
    LDB_C(0, aA, b0A, b1A, b2A, b3A);
    LDB_C(4, aB, b0B, b1B, b2B, b3B);

    for (int k = 0; k <= LQ_ - 16; k += 8) {
      v2f a = aA, b0 = b0A, b1 = b1A, b2 = b2A, b3 = b3A;
      LDB_C(k + 8, aA, b0A, b1A, b2A, b3A);
      acc0 = WMMA_F32(a, b0, acc0);
      acc1 = WMMA_F32(a, b1, acc1);
      acc2 = WMMA_F32(a, b2, acc2);
      acc3 = WMMA_F32(a, b3, acc3);
      v2f a_ = aB, c0 = b0B, c1 = b1B, c2 = b2B, c3 = b3B;
      LDB_C(k + 12, aB, b0B, b1B, b2B, b3B);
      acc0 = WMMA_F32(a_, c0, acc0);
      acc1 = WMMA_F32(a_, c1, acc1);
      acc2 = WMMA_F32(a_, c2, acc2);
      acc3 = WMMA_F32(a_, c3, acc3);
    }
    acc0 = WMMA_F32(aA, b0A, acc0);
    acc1 = WMMA_F32(aA, b1A, acc1);
    acc2 = WMMA_F32(aA, b2A, acc2);
    acc3 = WMMA_F32(aA, b3A, acc3);
    acc0 = WMMA_F32(aB, b0B, acc0);
    acc1 = WMMA_F32(aB, b1B, acc1);
    acc2 = WMMA_F32(aB, b2B, acc2);
    acc3 = WMMA_F32(aB, b3B, acc3);
#undef LDB_C

    float* Crow = AV + (size_t)(pt * 16 + half * 8) * H_ + idx;
#pragma unroll
    for (int j = 0; j < 4; ++j) {
      const v8f acc = (j == 0) ? acc0 : (j == 1) ? acc1 : (j == 2) ? acc2 : acc3;
      float* Cp = Crow + n0 + j * 16;
#pragma unroll
      for (int r = 0; r < 8; ++r) Cp[(size_t)r * H_] = acc[r];
    }
  }
}

// ---------------------------------------------------------------------------
extern "C" void kernel_launch(void* const* d_in, const int* in_sizes, int n_in,
                              void* d_out, int out_size, void* d_ws, size_t ws_size,
                              hipStream_t stream) {
  const float* proj_p  = (const float*)d_in[0];
  const float* proj_q  = (const float*)d_in[1];
  const int*   seq_len = (const int*)d_in[2];
  const float* W       = (const float*)d_in[3];
  const float* bias    = (const float*)d_in[4];
  float* out = (float*)d_out;

  float* trans_q = (float*)d_ws;                        // [B, LQ, H]
  float* att_vec = trans_q + (size_t)B_ * LQ_ * H_;     // [B, LP, H]

  const int M      = B_ * LQ_;                          // 8192
  const int blocks = (M / 64) * (H_ / 128);             // 128 * 6 = 768

  gemm_nt_bias_kernel<<<blocks, 256, 0, stream>>>(
      proj_q, W, bias, trans_q, M, H_, H_, 0);

  attn_softmax_kernel<<<dim3(LP_ / 32, B_), 256, 32 * LQ_ * sizeof(float), stream>>>(
      proj_p, proj_q, trans_q, seq_len, att_vec);

  gemm_nt_bias_kernel<<<blocks, 256, 0, stream>>>(
      att_vec, W, bias, out, M, H_, H_, 1);
}